// RegularizedFMNetV2_76785425318701
// MI455X (gfx1250) — compile-verified
//
#include <hip/hip_runtime.h>

#define K_EV   128
#define N_VERT 60000
#define C_FEAT 352
#define LAMBDA 0.001f
// GAMMA = 0.5 -> sqrtf

typedef __attribute__((ext_vector_type(2))) float v2f;
typedef __attribute__((ext_vector_type(8))) float v8f;

__device__ __forceinline__ v8f wmma_f32_k4(v2f a, v2f b, v8f c) {
    // D(16x16,f32) = A(16x4,f32) * B(4x16,f32) + C
    return __builtin_amdgcn_wmma_f32_16x16x4_f32(
        /*neg_a=*/false, a, /*neg_b=*/false, b,
        /*c_mod=*/(short)0, c, /*reuse_a=*/false, /*reuse_b=*/false);
}

// ---------------------------------------------------------------------------
// Kernel 1: spectral projection  A = Ex @ Fx,  B = Ey @ Fy   (128 x 352 each)
// grid = 2 * 8 * 22 blocks, 256 threads (8 waves). Each wave reduces a
// disjoint v-slice with V_WMMA_F32_16X16X4_F32, partials reduced via LDS.
// ---------------------------------------------------------------------------
__global__ __launch_bounds__(256) void proj_kernel(
    const float* __restrict__ evx, const float* __restrict__ fx,
    const float* __restrict__ evy, const float* __restrict__ fy,
    float* __restrict__ Aw, float* __restrict__ Bw)
{
    const int KT = K_EV / 16;     // 8
    const int CT = C_FEAT / 16;   // 22

    int bid = blockIdx.x;
    int mat = bid / (KT * CT);
    int rem = bid % (KT * CT);
    int k0  = (rem / CT) * 16;
    int c0  = (rem % CT) * 16;

    const float* E = mat ? evy : evx;   // (128, 60000) row-major
    const float* F = mat ? fy  : fx;    // (60000, 352) row-major
    float*       D = mat ? Bw  : Aw;    // (128, 352)

    int wave = threadIdx.x >> 5;
    int lane = threadIdx.x & 31;
    int m    = lane & 15;               // A: row M; B: col N
    int half = lane >> 4;               // selects K pair {0,1} vs {2,3}

    const int VS = N_VERT / 8;          // 7500, multiple of 4
    const int v0 = wave * VS;

    // A operand: evecs[k0+m][v + 2*half + {0,1}] -> contiguous b64
    const float* erow = E + (size_t)(k0 + m) * N_VERT + 2 * half;
    // B operand: feat[v + 2*half + j][c0 + m] -> two strided b32
    const float* f0 = F + (size_t)(2 * half) * C_FEAT + (c0 + m);
    const float* f1 = f0 + C_FEAT;

    v8f acc = {};
    for (int v = v0; v < v0 + VS; v += 4) {
        v2f a = *reinterpret_cast<const v2f*>(erow + v);
        v2f b;
        b.x = f0[(size_t)v * C_FEAT];
        b.y = f1[(size_t)v * C_FEAT];
        acc = wmma_f32_k4(a, b, acc);
    }

    // Reduce 8 wave-partials of the 16x16 tile through LDS.
    __shared__ float red[8][256];
#pragma unroll
    for (int r = 0; r < 8; ++r) red[wave][r * 32 + lane] = acc[r];
    __syncthreads();

    int t  = threadIdx.x;
    int rr = t >> 5;        // vgpr slot 0..7
    int lr = t & 31;        // lane 0..31
    float sum = 0.f;
#pragma unroll
    for (int w = 0; w < 8; ++w) sum += red[w][rr * 32 + lr];
    int row = k0 + rr + 8 * (lr >> 4);
    int col = c0 + (lr & 15);
    D[(size_t)row * C_FEAT + col] = sum;
}

// ---------------------------------------------------------------------------
// Kernel 2: M = B@B^T + C1@C1^T ;  rhs = A@B^T + C1^T    (both 128x128)
// 64 blocks x 1 wave; each wave owns one 16x16 tile. Every WMMA operand of
// X@Y^T is a contiguous b64 load (K runs along the row of the source).
// ---------------------------------------------------------------------------
__global__ __launch_bounds__(32) void gram_kernel(
    const float* __restrict__ Aw, const float* __restrict__ Bw,
    const float* __restrict__ C1,
    float* __restrict__ Mw, float* __restrict__ Rw)
{
    int k0 = (blockIdx.x >> 3) * 16;
    int l0 = (blockIdx.x & 7) * 16;
    int lane = threadIdx.x & 31;
    int m    = lane & 15;
    int half = lane >> 4;

    const float* bk = Bw + (size_t)(k0 + m) * C_FEAT + 2 * half;
    const float* bl = Bw + (size_t)(l0 + m) * C_FEAT + 2 * half;
    const float* ak = Aw + (size_t)(k0 + m) * C_FEAT + 2 * half;
    const float* ck = C1 + (size_t)(k0 + m) * K_EV + 2 * half;
    const float* cl = C1 + (size_t)(l0 + m) * K_EV + 2 * half;

    v8f accM = {};
    v8f accR = {};
    for (int c = 0; c < C_FEAT; c += 4) {       // 88 steps
        v2f vbl = *reinterpret_cast<const v2f*>(bl + c);   // B^T operand
        v2f vbk = *reinterpret_cast<const v2f*>(bk + c);
        v2f vak = *reinterpret_cast<const v2f*>(ak + c);
        accM = wmma_f32_k4(vbk, vbl, accM);      // BBt
        accR = wmma_f32_k4(vak, vbl, accR);      // ABt
    }
    for (int l = 0; l < K_EV; l += 4) {          // 32 steps
        v2f vck = *reinterpret_cast<const v2f*>(ck + l);
        v2f vcl = *reinterpret_cast<const v2f*>(cl + l);
        accM = wmma_f32_k4(vck, vcl, accM);      // CCt
    }

#pragma unroll
    for (int r = 0; r < 8; ++r) {
        int row = k0 + r + 8 * half;
        int col = l0 + m;
        Mw[row * K_EV + col] = accM[r];
        Rw[row * K_EV + col] = accR[r] + C1[col * K_EV + row];  // + C1^T
    }
}

// ---------------------------------------------------------------------------
// Kernel 3: for each i: solve (M + LAMBDA*diag(D[i,:])) x = rhs[i,:]
// System = Gram + nonneg diagonal -> SPD -> Gaussian elimination, no pivot.
// 128 blocks x 128 threads; S lives in LDS (128 x 129 pad, ~67 KB).
// ---------------------------------------------------------------------------
__global__ __launch_bounds__(128) void solve_kernel(
    const float* __restrict__ Mw, const float* __restrict__ Rw,
    const float* __restrict__ ex, const float* __restrict__ ey,
    float* __restrict__ out)
{
    __shared__ float S[K_EV][K_EV + 1];
    __shared__ float bvec[K_EV];
    __shared__ float xv[K_EV];
    __shared__ float sred[K_EV];

    const int t = threadIdx.x;
    const int i = blockIdx.x;

    // s = max(max evals_x, max evals_y)
    sred[t] = fmaxf(ex[t], ey[t]);
    __syncthreads();
    for (int off = 64; off > 0; off >>= 1) {
        if (t < off) sred[t] = fmaxf(sred[t], sred[t + off]);
        __syncthreads();
    }
    const float s = sred[0];

    // coalesced load M -> LDS
    for (int idx = t; idx < K_EV * K_EV; idx += 128)
        S[idx >> 7][idx & 127] = Mw[idx];
    __syncthreads();

    // diagonal regularizer D[i][t] (gamma = 0.5) and rhs row i
    {
        float ga = sqrtf(ey[i] / s);
        float gb = sqrtf(ex[t] / s);
        float ia = 1.f / (ga * ga + 1.f);
        float ib = 1.f / (gb * gb + 1.f);
        float re = ga * ia - gb * ib;
        float im = ia - ib;
        float d  = re * re + im * im;
        S[t][t] += LAMBDA * d;
        bvec[t] = Rw[i * K_EV + t];
    }

    // forward elimination (row-parallel)
    for (int j = 0; j < K_EV - 1; ++j) {
        __syncthreads();
        float pj = S[j][j];
        float bj = bvec[j];
        if (t > j) {
            float f = S[t][j] / pj;
            for (int c = j + 1; c < K_EV; ++c) S[t][c] -= f * S[j][c];
            bvec[t] -= f * bj;
        }
    }

    // back substitution
    for (int j = K_EV - 1; j >= 0; --j) {
        __syncthreads();
        if (t == j) xv[j] = bvec[j] / S[j][j];
        __syncthreads();
        if (t < j) bvec[t] -= S[t][j] * xv[j];
    }
    __syncthreads();
    out[(size_t)i * K_EV + t] = xv[t];   // C[0, i, t]
}

// ---------------------------------------------------------------------------
extern "C" void kernel_launch(void* const* d_in, const int* in_sizes, int n_in,
                              void* d_out, int out_size, void* d_ws, size_t ws_size,
                              hipStream_t stream) {
    const float* feat_x  = (const float*)d_in[0];
    const float* feat_y  = (const float*)d_in[1];
    const float* evals_x = (const float*)d_in[2];
    const float* evals_y = (const float*)d_in[3];
    const float* evt_x   = (const float*)d_in[4];
    const float* evt_y   = (const float*)d_in[5];
    const float* C1      = (const float*)d_in[6];

    float* ws = (float*)d_ws;
    float* Aw = ws;                      // 128*352
    float* Bw = Aw + K_EV * C_FEAT;      // 128*352
    float* Mw = Bw + K_EV * C_FEAT;      // 128*128
    float* Rw = Mw + K_EV * K_EV;        // 128*128

    proj_kernel<<<dim3(2 * (K_EV / 16) * (C_FEAT / 16)), dim3(256), 0, stream>>>(
        evt_x, feat_x, evt_y, feat_y, Aw, Bw);
    gram_kernel<<<dim3((K_EV / 16) * (K_EV / 16)), dim3(32), 0, stream>>>(
        Aw, Bw, C1, Mw, Rw);
    solve_kernel<<<dim3(K_EV), dim3(K_EV), 0, stream>>>(
        Mw, Rw, evals_x, evals_y, (float*)d_out);
}